// CELoss_51634096832929
// MI455X (gfx1250) — compile-verified
//
#include <hip/hip_runtime.h>
#include <hip/hip_bf16.h>
#include <stdint.h>

// Problem constants (from reference): input [8,14,512,512] f32, target [8,512,512] i64
#define C_CH    14
#define HW      262144          // 512*512
#define TILE    512             // pixels per tile (HW % TILE == 0)
#define TILES   4096            // NHW / TILE
#define NBLK    1024            // blocks
#define TPB_T   (TILES / NBLK)  // tiles per block = 4

#define EPS_A   (0.1f / 13.0f)  // cold smooth value
#define EPS_B   (0.9f - EPS_A)  // hot delta

typedef unsigned int v4u __attribute__((ext_vector_type(4)));
typedef int          v8i __attribute__((ext_vector_type(8)));
typedef int          v4i __attribute__((ext_vector_type(4)));

// ---------------------------------------------------------------------------
// TDM: one tensor_load_to_lds stages a full 14x512 f32 tile (28KB).
// 2D descriptor: x = 512 contiguous pixels, y = 14 channels, stride = HW.
// Wave-level op (EXEC ignored) -> issued by wave 0 only. Tracked by TENSORcnt.
// ---------------------------------------------------------------------------
__device__ __forceinline__ void tdm_issue(const float* __restrict__ inp,
                                          int tile, float* smbuf) {
  const int P0 = tile * TILE;
  const int n  = P0 >> 18;                 // / HW
  const int p0 = P0 & (HW - 1);            // % HW
  const uint64_t ga = (uint64_t)(uintptr_t)inp +
      (((uint64_t)(n * C_CH)) * (uint64_t)HW + (uint64_t)p0) * 4ull;
  const uint32_t la = (uint32_t)(uintptr_t)smbuf;   // LDS byte offset

  // D# group 0: count=1 | lds_addr | global_addr[56:0] | type=2 ("image")
  v4u g0;
  g0[0] = 1u;
  g0[1] = la;
  g0[2] = (uint32_t)ga;
  g0[3] = ((uint32_t)(ga >> 32) & 0x01FFFFFFu) | (2u << 30);

  // D# group 1: data_size=4B; tensor_dim0=HW, tensor_dim1=C; tile 512x14;
  //             tensor_dim0_stride=HW; no pad/iterate/multicast/barrier.
  v8i g1;
  g1[0] = 0x20000;                 // data_size=2 (4 bytes)
  g1[1] = 0;                       // barrier addr 0 | tensor_dim0[15:0]=0
  g1[2] = 0x000E0004;              // tensor_dim0[31:16]=4 | tensor_dim1=14
  g1[3] = (int)(512u << 16);       // tensor_dim1 hi=0 | tile_dim0=512
  g1[4] = 14;                      // tile_dim1=14 | tile_dim2=0
  g1[5] = 0x40000;                 // tensor_dim0_stride = HW
  g1[6] = 0;
  g1[7] = 0;

  v4i g2 = {0, 0, 0, 0};           // 2D tile: dims 2..4 unused
  v4i g3 = {0, 0, 0, 0};

#if __has_include(<hip/amd_detail/amd_gfx1250_TDM.h>)
  v8i g4 = {0, 0, 0, 0, 0, 0, 0, 0};
  __builtin_amdgcn_tensor_load_to_lds(g0, g1, g2, g3, g4, 0);
#else
  __builtin_amdgcn_tensor_load_to_lds(g0, g1, g2, g3, 0);
#endif
}

// ---------------------------------------------------------------------------
// Async global->LDS staging of the int64 targets for one tile (4KB):
// 256 lanes x one global_load_async_to_lds_b128 (2 targets per lane).
// ---------------------------------------------------------------------------
__device__ __forceinline__ void issue_tgt(const long long* __restrict__ tgt,
                                          int tile, long long* tb, int tid) {
  const uint64_t ga = (uint64_t)(uintptr_t)(tgt + tile * TILE) + (uint64_t)(tid * 16);
  const uint32_t la = (uint32_t)(uintptr_t)tb + (uint32_t)(tid * 16);
  asm volatile("global_load_async_to_lds_b128 %0, %1, off"
               :: "v"(la), "v"(ga)
               : "memory");
}

// ---------------------------------------------------------------------------
// Consume one tile from LDS: per-pixel smoothed-CE math in closed form.
// ---------------------------------------------------------------------------
__device__ __forceinline__ void compute_tile(const long long* __restrict__ tb,
                                             const float* __restrict__ smbuf,
                                             int tid, float& loss_sel, float& sw_sel,
                                             float& sw_total) {
#pragma unroll
  for (int h = 0; h < 2; ++h) {
    const int p = tid + h * 256;
    const int t = ((const int*)tb)[2 * p];   // low dword of int64 target

    float x[C_CH];
    float m = -3.0e38f, S = 0.0f, xt = 0.0f;
#pragma unroll
    for (int c = 0; c < C_CH; ++c) {
      const float v = smbuf[c * TILE + p];
      x[c] = v;
      m = fmaxf(m, v);
      S += v;
      if (c == t) xt = v;                    // cndmask chain, no dyn reg index
    }
    float se = 0.0f;
#pragma unroll
    for (int c = 0; c < C_CH; ++c) se += __expf(x[c] - m);
    const float lse = m + __logf(se);

    const bool  t0      = (t == 0);
    const float sum_key = t0 ? -0.8f : (0.9f + 11.0f * EPS_A);
    const float sgn     = t0 ? -1.0f : 1.0f;
    const float dot     = EPS_A * (S - 2.0f * x[0]) + EPS_B * sgn * xt;
    const float loss    = lse * sum_key - dot;
    const float sws     = t0 ? 0.1f : (0.9f + 12.0f * EPS_A);

    const bool pos = loss > 0.0f;
    loss_sel += pos ? loss : 0.0f;
    sw_sel   += pos ? sws  : 0.0f;
    sw_total += sws;
  }
}

// ---------------------------------------------------------------------------
// Main kernel: TDM (input) + async-LDS (targets), double-buffered, 4 tiles/blk.
// ---------------------------------------------------------------------------
__global__ __launch_bounds__(256) void ce_main(const float* __restrict__ inp,
                                               const long long* __restrict__ tgt,
                                               float* __restrict__ ws) {
  __shared__ float     sm[2][C_CH * TILE];   // 2 x 28KB input tiles
  __shared__ long long tg[2][TILE];          // 2 x 4KB target tiles

  const int tid  = threadIdx.x;
  const int bid  = blockIdx.x;
  const int wave = tid >> 5;

  float ls = 0.0f, ss = 0.0f, st = 0.0f;

  issue_tgt(tgt, bid, tg[0], tid);           // prologue
  if (wave == 0) tdm_issue(inp, bid, sm[0]);

  for (int i = 0; i < TPB_T; ++i) {
    const int buf = i & 1;
    if (i + 1 < TPB_T) {
      issue_tgt(tgt, bid + (i + 1) * NBLK, tg[buf ^ 1], tid);
      if (wave == 0) {
        tdm_issue(inp, bid + (i + 1) * NBLK, sm[buf ^ 1]);
        // 1 new TDM op outstanding; in-order per wave => <=1 means tile i done
        __builtin_amdgcn_s_wait_tensorcnt(1);
      }
      asm volatile("s_wait_asynccnt 0x1" ::: "memory");
    } else {
      if (wave == 0) __builtin_amdgcn_s_wait_tensorcnt(0);
      asm volatile("s_wait_asynccnt 0x0" ::: "memory");
    }
    __syncthreads();                         // tile visible block-wide
    compute_tile(tg[buf], sm[buf], tid, ls, ss, st);
    __syncthreads();                         // buffer reusable next iteration
  }

  // wave32 shuffle reduction, then one atomic triple per wave
#pragma unroll
  for (int off = 16; off > 0; off >>= 1) {
    ls += __shfl_down(ls, off, 32);
    ss += __shfl_down(ss, off, 32);
    st += __shfl_down(st, off, 32);
  }
  if ((tid & 31) == 0) {
    atomicAdd(&ws[0], ls);
    atomicAdd(&ws[1], ss);
    atomicAdd(&ws[2], st);
  }
}

__global__ void ce_init(float* __restrict__ ws) {
  if (threadIdx.x < 3) ws[threadIdx.x] = 0.0f;
}

__global__ void ce_fin(const float* __restrict__ ws, float* __restrict__ out) {
  const float ls = ws[0], ss = ws[1], st = ws[2];
  const float denom = (ss == 0.0f) ? 1.0f : ss;
  out[0] = (st == 0.0f) ? ls : ls / denom;
}

extern "C" void kernel_launch(void* const* d_in, const int* in_sizes, int n_in,
                              void* d_out, int out_size, void* d_ws, size_t ws_size,
                              hipStream_t stream) {
  (void)in_sizes; (void)n_in; (void)out_size; (void)ws_size;
  const float*     inp = (const float*)d_in[0];
  const long long* tgt = (const long long*)d_in[1];
  float*           out = (float*)d_out;
  float*           ws  = (float*)d_ws;

  ce_init<<<1, 32, 0, stream>>>(ws);
  ce_main<<<NBLK, 256, 0, stream>>>(inp, tgt, ws);
  ce_fin<<<1, 1, 0, stream>>>(ws, out);
}